// PhysicsGNN_NC_66305705116250
// MI455X (gfx1250) — compile-verified
//
#include <hip/hip_runtime.h>

#define H_DIM   128
#define F_IN_DIM 256
#define C_DIM   40
#define L_DIM   4
#define STEPF   0.1f

typedef __attribute__((ext_vector_type(2))) float v2f;
typedef __attribute__((ext_vector_type(4))) float v4f;
typedef __attribute__((ext_vector_type(8))) float v8f;

// ---------------- setup kernels ----------------

__global__ void zero_two(float* __restrict__ a, float* __restrict__ b, int n) {
    int i = blockIdx.x * blockDim.x + threadIdx.x;
    if (i < n) { a[i] = 0.f; b[i] = 0.f; }
}

__global__ void deg_accum(const int* __restrict__ row, float* __restrict__ deg, int E) {
    int e = blockIdx.x * blockDim.x + threadIdx.x;
    if (e < E) atomicAdd(&deg[row[e]], 1.0f);
}

__global__ void deg_inv_kernel(float* __restrict__ deg, int n) {
    int i = blockIdx.x * blockDim.x + threadIdx.x;
    if (i < n) { float d = deg[i]; deg[i] = (d > 0.f) ? rsqrtf(d) : 0.f; }
}

__global__ void coef_accum(const int* __restrict__ row, const int* __restrict__ col,
                           const float* __restrict__ dinv, float* __restrict__ coef, int E) {
    int e = blockIdx.x * blockDim.x + threadIdx.x;
    if (e < E) {
        int r = row[e], c = col[e];
        atomicAdd(&coef[c], dinv[r] * dinv[c]);
    }
}

// W_eff[l] = W0 + W0^T + diag(Wr[:,-2]*sum|W0| + Wr[:,-1]);  one block per layer,
// thread i builds row i (symmetric, so W_eff == W_eff^T for the GEMMs).
__global__ void weff_kernel(const float* __restrict__ Wraw, float* __restrict__ Weff) {
    int l = blockIdx.x;
    int i = threadIdx.x;                      // 0..H-1
    const float* Wr = Wraw + (size_t)l * H_DIM * (H_DIM + 2);
    float*       We = Weff + (size_t)l * H_DIM * H_DIM;
    float s = 0.f;
    for (int j = 0; j < H_DIM; ++j) {
        float w = 0.f;
        if (j > i)      w = Wr[i * (H_DIM + 2) + j];
        else if (j < i) w = Wr[j * (H_DIM + 2) + i];
        s += fabsf(w);
        We[i * H_DIM + j] = w;
    }
    float d = Wr[i * (H_DIM + 2) + H_DIM] * s + Wr[i * (H_DIM + 2) + H_DIM + 1];
    We[i * H_DIM + i] = d;
}

// ---------------- WMMA GEMM kernels ----------------
// Operand packing for V_WMMA_F32_16X16X4_F32 (wave32):
//   A (16x4 MxK):  lane m = lane&15, k = k0 + 2*(lane>>4) + v   (v = vgpr 0/1)
//   B (4x16 KxN):  lane n = lane&15, k = k0 + 2*(lane>>4) + v
//   C (16x16):     row = r + 8*(lane>>4), col = lane&15   (r = vgpr 0..7)

// h[n, j] = sum_k x[n,k] * enc_W[j,k] + enc_b[j]
// x is 102 MB streamed exactly once -> non-temporal loads so the h buffers
// (3 x 51 MB, all layers) stay resident in the 192 MB L2.
__launch_bounds__(256)
__global__ void encoder_gemm(const float* __restrict__ x, const float* __restrict__ W,
                             const float* __restrict__ bias, float* __restrict__ h, int Nn) {
    __shared__ __align__(16) float As[16][F_IN_DIM + 4];
    const int row0 = blockIdx.x * 16;
    const v4f* x4 = (const v4f*)x;
    for (int idx = threadIdx.x; idx < 16 * (F_IN_DIM / 4); idx += 256) {
        int r  = idx >> 6;                 // / 64 float4s per row
        int c4 = idx & 63;
        int node = row0 + r;
        v4f v = {0.f, 0.f, 0.f, 0.f};
        if (node < Nn)
            v = __builtin_nontemporal_load(&x4[(size_t)node * (F_IN_DIM / 4) + c4]);
        *(v4f*)&As[r][c4 * 4] = v;         // ds_store_b128
    }
    __syncthreads();

    const int lane = threadIdx.x & 31;
    const int wave = threadIdx.x >> 5;    // 8 waves -> 8 x 16 cols = 128
    const int col0 = wave * 16;
    const int m    = lane & 15;
    const int hi   = lane >> 4;
    const int colN = col0 + m;
    const float* wrow = W + (size_t)colN * F_IN_DIM;   // B lane walks one enc_W row

    v8f acc = {};
    for (int k0 = 0; k0 < F_IN_DIM; k0 += 4) {
        int k = k0 + 2 * hi;
        v2f a = *(const v2f*)&As[m][k];
        v2f b = *(const v2f*)&wrow[k];
        acc = __builtin_amdgcn_wmma_f32_16x16x4_f32(false, a, false, b,
                                                    (short)0, acc, false, false);
    }
    const float bv = bias[colN];
    for (int r = 0; r < 8; ++r) {
        int node = row0 + r + 8 * hi;
        if (node < Nn) h[(size_t)node * H_DIM + colN] = acc[r] + bv;
    }
}

// out_p = h_in @ W_eff + W_b;  h_out = h_in + STEP*relu(coef*out_p - (h_in*ext_w + beta*h0))
__launch_bounds__(256)
__global__ void layer_gemm(const float* __restrict__ h_in, const float* __restrict__ h0,
                           const float* __restrict__ coef, const float* __restrict__ Weff,
                           const float* __restrict__ Wb, const float* __restrict__ extw,
                           const float* __restrict__ beta_p, float* __restrict__ h_out, int Nn) {
    __shared__ __align__(16) float As[16][H_DIM + 4];
    const int row0 = blockIdx.x * 16;
    const v4f* h4 = (const v4f*)h_in;
    for (int idx = threadIdx.x; idx < 16 * (H_DIM / 4); idx += 256) {
        int r  = idx >> 5;                 // / 32 float4s per row
        int c4 = idx & 31;
        int node = row0 + r;
        v4f v = {0.f, 0.f, 0.f, 0.f};
        if (node < Nn) v = h4[(size_t)node * (H_DIM / 4) + c4];
        *(v4f*)&As[r][c4 * 4] = v;
    }
    __syncthreads();

    const int lane = threadIdx.x & 31;
    const int wave = threadIdx.x >> 5;
    const int col0 = wave * 16;
    const int m    = lane & 15;
    const int hi   = lane >> 4;
    const int colN = col0 + m;

    v8f acc = {};
    for (int k0 = 0; k0 < H_DIM; k0 += 4) {
        int k = k0 + 2 * hi;
        v2f a = *(const v2f*)&As[m][k];
        v2f b;
        b.x = Weff[(size_t)k * H_DIM + colN];        // coalesced 16-lane rows
        b.y = Weff[(size_t)(k + 1) * H_DIM + colN];
        acc = __builtin_amdgcn_wmma_f32_16x16x4_f32(false, a, false, b,
                                                    (short)0, acc, false, false);
    }

    const float wbv = Wb[colN];
    const float ew  = extw[colN];
    const float bl  = *beta_p;
    for (int r = 0; r < 8; ++r) {
        int node = row0 + r + 8 * hi;
        if (node < Nn) {
            float cf   = coef[node];
            float hv   = As[r + 8 * hi][colN];       // h_in from LDS tile
            float h0v  = h0[(size_t)node * H_DIM + colN];
            float outp = acc[r] + wbv;
            float outv = cf * outp - (hv * ew + bl * h0v);
            h_out[(size_t)node * H_DIM + colN] = hv + STEPF * fmaxf(outv, 0.f);
        }
    }
}

// out[n, c] = sum_k h[n,k] * dec_W[c,k] + dec_b[c];  C=40 -> 3 col-tiles (3 waves), guarded.
// Output is written once and never re-read on device -> non-temporal stores.
__launch_bounds__(96)
__global__ void decoder_gemm(const float* __restrict__ h, const float* __restrict__ W,
                             const float* __restrict__ bias, float* __restrict__ out, int Nn) {
    __shared__ __align__(16) float As[16][H_DIM + 4];
    const int row0 = blockIdx.x * 16;
    const v4f* h4 = (const v4f*)h;
    for (int idx = threadIdx.x; idx < 16 * (H_DIM / 4); idx += 96) {
        int r  = idx >> 5;
        int c4 = idx & 31;
        int node = row0 + r;
        v4f v = {0.f, 0.f, 0.f, 0.f};
        if (node < Nn) v = h4[(size_t)node * (H_DIM / 4) + c4];
        *(v4f*)&As[r][c4 * 4] = v;
    }
    __syncthreads();

    const int lane = threadIdx.x & 31;
    const int wave = threadIdx.x >> 5;    // 0..2
    const int col0 = wave * 16;
    const int m    = lane & 15;
    const int hi   = lane >> 4;
    const int colN = col0 + m;
    const bool colok = (colN < C_DIM);
    const float* wrow = W + (size_t)(colok ? colN : 0) * H_DIM;

    v8f acc = {};
    for (int k0 = 0; k0 < H_DIM; k0 += 4) {
        int k = k0 + 2 * hi;
        v2f a = *(const v2f*)&As[m][k];
        v2f b = {0.f, 0.f};
        if (colok) b = *(const v2f*)&wrow[k];
        acc = __builtin_amdgcn_wmma_f32_16x16x4_f32(false, a, false, b,
                                                    (short)0, acc, false, false);
    }
    if (colok) {
        const float bv = bias[colN];
        for (int r = 0; r < 8; ++r) {
            int node = row0 + r + 8 * hi;
            if (node < Nn)
                __builtin_nontemporal_store(acc[r] + bv,
                                            &out[(size_t)node * C_DIM + colN]);
        }
    }
}

// ---------------- host launcher ----------------

extern "C" void kernel_launch(void* const* d_in, const int* in_sizes, int n_in,
                              void* d_out, int out_size, void* d_ws, size_t ws_size,
                              hipStream_t stream) {
    const float* x     = (const float*)d_in[0];
    const int*   eidx  = (const int*)d_in[1];
    const float* enc_W = (const float*)d_in[2];
    const float* enc_b = (const float*)d_in[3];
    const float* W_raw = (const float*)d_in[4];
    const float* W_b   = (const float*)d_in[5];
    const float* ext_w = (const float*)d_in[6];
    const float* beta  = (const float*)d_in[7];
    const float* dec_W = (const float*)d_in[8];
    const float* dec_b = (const float*)d_in[9];

    const int Nn = in_sizes[0] / F_IN_DIM;
    const int E  = in_sizes[1] / 2;
    const int* row = eidx;
    const int* col = eidx + E;

    char* ws = (char*)d_ws;
    size_t off = 0;
    auto wsalloc = [&](size_t bytes) -> void* {
        void* p = ws + off;
        off = (off + bytes + 255) & ~(size_t)255;
        return p;
    };
    float* deg  = (float*)wsalloc((size_t)Nn * sizeof(float));            // becomes deg_inv in place
    float* coef = (float*)wsalloc((size_t)Nn * sizeof(float));
    float* Weff = (float*)wsalloc((size_t)L_DIM * H_DIM * H_DIM * sizeof(float));
    float* h0   = (float*)wsalloc((size_t)Nn * H_DIM * sizeof(float));
    float* ha   = (float*)wsalloc((size_t)Nn * H_DIM * sizeof(float));
    float* hb   = (float*)wsalloc((size_t)Nn * H_DIM * sizeof(float));
    (void)ws_size; (void)n_in; (void)out_size;

    const int tb = 256;
    zero_two<<<(Nn + tb - 1) / tb, tb, 0, stream>>>(deg, coef, Nn);
    deg_accum<<<(E + tb - 1) / tb, tb, 0, stream>>>(row, deg, E);
    deg_inv_kernel<<<(Nn + tb - 1) / tb, tb, 0, stream>>>(deg, Nn);
    coef_accum<<<(E + tb - 1) / tb, tb, 0, stream>>>(row, col, deg, coef, E);
    weff_kernel<<<L_DIM, H_DIM, 0, stream>>>(W_raw, Weff);

    const int gTiles = (Nn + 15) / 16;
    encoder_gemm<<<gTiles, 256, 0, stream>>>(x, enc_W, enc_b, h0, Nn);

    const float* hin = h0;
    float* bufs[2] = { ha, hb };
    for (int l = 0; l < L_DIM; ++l) {
        float* hout = bufs[l & 1];
        layer_gemm<<<gTiles, 256, 0, stream>>>(hin, h0, coef,
                                               Weff + (size_t)l * H_DIM * H_DIM,
                                               W_b + (size_t)l * H_DIM,
                                               ext_w + (size_t)l * H_DIM,
                                               beta + l, hout, Nn);
        hin = hout;
    }
    decoder_gemm<<<gTiles, 96, 0, stream>>>(hin, dec_W, dec_b, (float*)d_out, Nn);
}